// MPHeteHead_14448269984047
// MI455X (gfx1250) — compile-verified
//
#include <hip/hip_runtime.h>
#include <hip/hip_bf16.h>
#include <math.h>

typedef __attribute__((ext_vector_type(2))) float v2f;
typedef __attribute__((ext_vector_type(8))) float v8f;

#define LDS_STRIDE 130  // pad 128 -> 130 so 16 lanes reading col k of rows 0..15 hit distinct banks

// ---------------------------------------------------------------------------
// Row-wise L2 normalize: out[r] = in[r] / max(||in[r]||, 1e-12). One block per
// row, 128 threads (4 waves), wave32 shfl reduction + tiny LDS combine.
// ---------------------------------------------------------------------------
__global__ __launch_bounds__(128) void l2norm_rows(const float* __restrict__ in,
                                                   float* __restrict__ out) {
  const size_t row = blockIdx.x;
  const int t = threadIdx.x;
  __shared__ float wsum[4];
  float v = in[row * 128 + t];
  float s = v * v;
#pragma unroll
  for (int off = 16; off; off >>= 1) s += __shfl_xor(s, off, 32);
  if ((t & 31) == 0) wsum[t >> 5] = s;
  __syncthreads();
  if (t == 0) {
    float tot = wsum[0] + wsum[1] + wsum[2] + wsum[3];
    wsum[0] = 1.0f / fmaxf(sqrtf(tot), 1e-12f);
  }
  __syncthreads();
  out[row * 128 + t] = v * wsum[0];
}

// ---------------------------------------------------------------------------
// Fused bipartite edge aggregation. One wave per edge; lane holds 4 floats
// (float4). Gathers x_data[src_d]*ev -> atomic add into agg_t[src_t] and
// x_task[src_t]*ev -> atomic add into agg_d[src_d].
// ---------------------------------------------------------------------------
__global__ __launch_bounds__(256) void edge_agg(
    const float* __restrict__ xd, const float* __restrict__ xt,
    const float* __restrict__ ev, const int* __restrict__ src_d,
    const int* __restrict__ src_t, float* __restrict__ agg_t,
    float* __restrict__ agg_d, int E) {
  const int e = blockIdx.x * 8 + (threadIdx.x >> 5);
  if (e >= E) return;
  const int lane = threadIdx.x & 31;
  const int d = src_d[e];
  const int t = src_t[e];
  const float w = ev[e];
  const float4 vd = *(const float4*)(xd + (size_t)d * 128 + lane * 4);
  const float4 vt = *(const float4*)(xt + (size_t)t * 128 + lane * 4);
  float* at = agg_t + (size_t)t * 128 + lane * 4;
  float* ad = agg_d + (size_t)d * 128 + lane * 4;
  atomicAdd(at + 0, vd.x * w);
  atomicAdd(at + 1, vd.y * w);
  atomicAdd(at + 2, vd.z * w);
  atomicAdd(at + 3, vd.w * w);
  atomicAdd(ad + 0, vt.x * w);
  atomicAdd(ad + 1, vt.y * w);
  atomicAdd(ad + 2, vt.z * w);
  atomicAdd(ad + 3, vt.w * w);
}

// ---------------------------------------------------------------------------
// Fused H = A @ W0 + X @ W1 ; optional relu ; row l2norm ; M x 128, W 128x128.
// Block = 256 threads = 8 waves. Block owns 16 output rows x 128 cols; wave w
// owns column tile [16w, 16w+16). A/X 16x128 tiles staged in LDS (padded
// stride). The two GEMM terms run in INDEPENDENT accumulators interleaved in
// one loop -> two parallel v_wmma_f32_16x16x4_f32 dependency chains, so loads
// of one chain hide under math of the other. Requires M % 16 == 0
// (holds: 100000 = 16*6250, 256 = 16*16).
// ---------------------------------------------------------------------------
__global__ __launch_bounds__(256) void gemm2_relu_norm(
    const float* __restrict__ A, const float* __restrict__ W0,
    const float* __restrict__ X, const float* __restrict__ W1,
    float* __restrict__ out, int relu) {
  __shared__ float ldsA[16 * LDS_STRIDE];
  __shared__ float ldsX[16 * LDS_STRIDE];
  __shared__ float rscale[16];
  const int tid = threadIdx.x;
  const int lane = tid & 31;
  const int wave = tid >> 5;
  const size_t row0 = (size_t)blockIdx.x * 16;

  // Stage A and X tiles (16x128 each) into LDS, coalesced 8 floats/thread.
#pragma unroll
  for (int i = 0; i < 8; ++i) {
    int e = tid * 8 + i;  // 0..2047
    int r = e >> 7, c = e & 127;
    ldsA[r * LDS_STRIDE + c] = A[(row0 + r) * 128 + c];
    ldsX[r * LDS_STRIDE + c] = X[(row0 + r) * 128 + c];
  }
  __syncthreads();

  const int m = lane & 15;          // A row / B col within tile
  const int kh = (lane >> 4) << 1;  // K sub-offset: 0 or 2
  const int n0 = wave * 16;         // this wave's output column tile
  v8f acc0 = {};
  v8f acc1 = {};
  for (int k = 0; k < 128; k += 4) {
    v2f a0 = *(const v2f*)&ldsA[m * LDS_STRIDE + k + kh];
    v2f b0;
    b0.x = W0[(size_t)(k + kh) * 128 + n0 + m];
    b0.y = W0[(size_t)(k + kh + 1) * 128 + n0 + m];
    acc0 = __builtin_amdgcn_wmma_f32_16x16x4_f32(false, a0, false, b0, (short)0,
                                                 acc0, false, false);
    v2f a1 = *(const v2f*)&ldsX[m * LDS_STRIDE + k + kh];
    v2f b1;
    b1.x = W1[(size_t)(k + kh) * 128 + n0 + m];
    b1.y = W1[(size_t)(k + kh + 1) * 128 + n0 + m];
    acc1 = __builtin_amdgcn_wmma_f32_16x16x4_f32(false, a1, false, b1, (short)0,
                                                 acc1, false, false);
  }
  __syncthreads();  // done reading ldsA/ldsX; reuse ldsA for the output tile

  // Scatter D fragment (+relu) to LDS: lane l, vgpr v -> row v+(l>>4)*8,
  // col n0+(l&15).
#pragma unroll
  for (int v = 0; v < 8; ++v) {
    int r = v + ((lane >> 4) << 3);
    float val = acc0[v] + acc1[v];
    if (relu) val = fmaxf(val, 0.0f);
    ldsA[r * LDS_STRIDE + n0 + m] = val;
  }
  __syncthreads();

  // Row l2 norms: thread tid -> (row = tid>>4, seg = tid&15), 8 elems each,
  // 16-lane shfl reduce.
  {
    const int r = tid >> 4, seg = tid & 15;
    float s = 0.0f;
#pragma unroll
    for (int j = 0; j < 8; ++j) {
      float v = ldsA[r * LDS_STRIDE + seg * 8 + j];
      s += v * v;
    }
#pragma unroll
    for (int off = 8; off; off >>= 1) s += __shfl_xor(s, off, 16);
    if (seg == 0) rscale[r] = 1.0f / fmaxf(sqrtf(s), 1e-12f);
  }
  __syncthreads();
  {
    const int r = tid >> 4, seg = tid & 15;
    const float sc = rscale[r];
#pragma unroll
    for (int j = 0; j < 8; ++j)
      out[(row0 + r) * 128 + seg * 8 + j] =
          ldsA[r * LDS_STRIDE + seg * 8 + j] * sc;
  }
}

// ---------------------------------------------------------------------------
// pred = XD (M x 128) @ XT^T (128 x 256). Block = 512 threads = 16 waves; wave
// w owns columns [16w, 16w+16). Split-K: k and k+64 halves accumulate in
// independent v8f accumulators (two parallel WMMA chains), summed at the end.
// B fragments are contiguous float2 loads from row-major XT (B = XT^T); XT is
// 128KB -> L2-resident.
// ---------------------------------------------------------------------------
__global__ __launch_bounds__(512) void pred_gemm(const float* __restrict__ XD,
                                                 const float* __restrict__ XT,
                                                 float* __restrict__ out) {
  __shared__ float ldsA[16 * LDS_STRIDE];
  const int tid = threadIdx.x;
  const int lane = tid & 31;
  const int wave = tid >> 5;  // 0..15
  const size_t row0 = (size_t)blockIdx.x * 16;
  for (int i = tid; i < 16 * 128; i += 512) {
    int r = i >> 7, c = i & 127;
    ldsA[r * LDS_STRIDE + c] = XD[(row0 + r) * 128 + c];
  }
  __syncthreads();
  const int m = lane & 15;
  const int kh = (lane >> 4) << 1;
  const int n0 = wave * 16;
  const float* Brow = XT + (size_t)(n0 + m) * 128;
  v8f acc0 = {};
  v8f acc1 = {};
  for (int k = 0; k < 64; k += 4) {
    v2f a0 = *(const v2f*)&ldsA[m * LDS_STRIDE + k + kh];
    v2f b0 = *(const v2f*)&Brow[k + kh];
    acc0 = __builtin_amdgcn_wmma_f32_16x16x4_f32(false, a0, false, b0, (short)0,
                                                 acc0, false, false);
    v2f a1 = *(const v2f*)&ldsA[m * LDS_STRIDE + k + 64 + kh];
    v2f b1 = *(const v2f*)&Brow[k + 64 + kh];
    acc1 = __builtin_amdgcn_wmma_f32_16x16x4_f32(false, a1, false, b1, (short)0,
                                                 acc1, false, false);
  }
#pragma unroll
  for (int v = 0; v < 8; ++v) {
    int r = v + ((lane >> 4) << 3);
    out[(row0 + r) * 256 + n0 + m] = acc0[v] + acc1[v];
  }
}

// ---------------------------------------------------------------------------
// Host-side orchestration (graph-capture safe: only async launches/memsets).
// ---------------------------------------------------------------------------
extern "C" void kernel_launch(void* const* d_in, const int* in_sizes, int n_in,
                              void* d_out, int out_size, void* d_ws,
                              size_t ws_size, hipStream_t stream) {
  const float* gf   = (const float*)d_in[0];  // graph_feature [Nd,128]
  const float* ev   = (const float*)d_in[1];  // edge values  [E]
  const float* te   = (const float*)d_in[2];  // task_emb     [Nt,128]
  const float* Wd2t = (const float*)d_in[3];  // [2,128,128]
  const float* Wt2d = (const float*)d_in[4];  // [2,128,128]
  const float* Wsd  = (const float*)d_in[5];  // W_self_data [2,128,128]
  const float* Wst  = (const float*)d_in[6];  // W_self_task [2,128,128]
  const int* srcd   = (const int*)d_in[7];    // [E]
  const int* srct   = (const int*)d_in[8];    // [E]

  const int D = 128;
  const int Nd = in_sizes[0] / D;  // 100000
  const int E  = in_sizes[1];      // 800000
  const int Nt = in_sizes[2] / D;  // 256

  float* out        = (float*)d_out;
  float* pred       = out;                        // [Nd, Nt]
  float* data_emb   = out + (size_t)Nd * Nt;      // [Nd, D]
  float* task_emb_n = data_emb + (size_t)Nd * D;  // [Nt, D]

  float* ws = (float*)d_ws;
  size_t o = 0;
  float* agg_d = ws + o; o += (size_t)Nd * D;
  float* xd1   = ws + o; o += (size_t)Nd * D;
  float* xd2   = ws + o; o += (size_t)Nd * D;
  float* agg_t = ws + o; o += (size_t)Nt * D;
  float* xt1   = ws + o; o += (size_t)Nt * D;
  float* xt2   = ws + o; o += (size_t)Nt * D;

  // Normalized inputs are both outputs (data_emb/task_emb_n) and layer-0
  // activations.
  l2norm_rows<<<Nd, 128, 0, stream>>>(gf, data_emb);
  l2norm_rows<<<Nt, 128, 0, stream>>>(te, task_emb_n);

  const float* xd = data_emb;
  const float* xt = task_emb_n;
  float* xdn[2] = {xd1, xd2};
  float* xtn[2] = {xt1, xt2};

  for (int l = 0; l < 2; ++l) {
    hipMemsetAsync(agg_d, 0, (size_t)Nd * D * sizeof(float), stream);
    hipMemsetAsync(agg_t, 0, (size_t)Nt * D * sizeof(float), stream);
    edge_agg<<<(E + 7) / 8, 256, 0, stream>>>(xd, xt, ev, srcd, srct, agg_t,
                                              agg_d, E);
    const size_t wl = (size_t)l * D * D;
    gemm2_relu_norm<<<Nt / 16, 256, 0, stream>>>(agg_t, Wd2t + wl, xt,
                                                 Wst + wl, xtn[l], l == 0);
    gemm2_relu_norm<<<Nd / 16, 256, 0, stream>>>(agg_d, Wt2d + wl, xd,
                                                 Wsd + wl, xdn[l], l == 0);
    xd = xdn[l];
    xt = xtn[l];
  }

  pred_gemm<<<Nd / 16, 512, 0, stream>>>(xd, xt, pred);
}